// NMS_35845797052444
// MI455X (gfx1250) — compile-verified
//
#include <hip/hip_runtime.h>
#include <hip/hip_bf16.h>
#include <stdint.h>

typedef __attribute__((ext_vector_type(2))) float v2f;
typedef __attribute__((ext_vector_type(8))) float v8f;

#define N_TOP       4096
#define CAND_CAP    8192
#define CONF_THRESH 0.5f
#define IOU_THRESH  0.6f
#define MASK_W      128      /* 4096 / 32 bits */

/* ---- workspace layout (bytes) ---- */
#define OFF_HIST   0u        /* 4096 u32 histogram over score-key bits [22:11]      */
#define OFF_CNT    16384u    /* 1 u32 candidate counter                             */
#define OFF_TBIN   16388u    /* 1 u32 threshold bin                                 */
#define OFF_CAND   16640u    /* 8192 u64 composite keys                             */
#define OFF_SCORE  82432u    /* 4096 f32 top scores (sorted desc)                   */
#define OFF_IDX    98816u    /* 4096 u32 original indices                           */
#define OFF_BOXES  115200u   /* 4096 float4 gathered boxes                          */
#define OFF_AREA   180736u   /* 4096 f32 areas                                      */
#define OFF_KEEP   197120u   /* 4096 u32 keep flags                                 */
#define OFF_MASK   213760u   /* 4096*128 u32 suppression bitmask (2 MB)             */

typedef unsigned int u32;
typedef unsigned long long u64;

/* ---------- pass 0: zero histogram + counters ---------- */
__global__ void zero_kernel(u32* hist_and_cnt) {
    int i = blockIdx.x * blockDim.x + threadIdx.x;
    if (i < 4096 + 2) hist_and_cnt[i] = 0u;
}

/* ---------- pass 1: histogram of valid-score key bits ---------- */
__global__ void hist_kernel(const float* __restrict__ confs, u32* __restrict__ hist, int n) {
    int i = blockIdx.x * blockDim.x + threadIdx.x;
    if (i >= n) return;
    float c = confs[i];
    if (c >= CONF_THRESH) {
        u32 key = __float_as_uint(c);          /* positive floats: bit order == value order */
        u32 bin = (key >> 11) & 4095u;         /* valid range [0.5,1): mantissa spreads bins */
        atomicAdd(&hist[bin], 1u);
    }
}

/* ---------- pass 2: find threshold bin (serial scan, trivial work) ---------- */
__global__ void scan_kernel(const u32* __restrict__ hist, u32* __restrict__ tbin) {
    if (threadIdx.x != 0 || blockIdx.x != 0) return;
    u32 running = 0, T = 0;
    for (int b = 4095; b >= 0; --b) {
        running += hist[b];
        if (running >= (u32)N_TOP) { T = (u32)b; break; }
    }
    *tbin = T;
}

/* ---------- pass 3: compact candidates with bin >= threshold ---------- */
__global__ void compact_kernel(const float* __restrict__ confs, const u32* __restrict__ tbin,
                               u32* __restrict__ cnt, u64* __restrict__ cand, int n) {
    int i = blockIdx.x * blockDim.x + threadIdx.x;
    if (i >= n) return;
    float c = confs[i];
    if (c < CONF_THRESH) return;
    u32 key = __float_as_uint(c);
    u32 bin = (key >> 11) & 4095u;
    if (bin < *tbin) return;
    u32 pos = atomicAdd(cnt, 1u);
    if (pos < (u32)CAND_CAP)
        /* composite: score bits desc, then original index asc on ties */
        cand[pos] = ((u64)key << 32) | (u64)(0xFFFFFFFFu - (u32)i);
}

/* ---------- pass 4: single-WG bitonic sort (8192 keys in 64KB LDS), gather top-4096 ---------- */
__global__ __launch_bounds__(1024)
void sort_gather_kernel(const float4* __restrict__ boxes, const u64* __restrict__ cand,
                        const u32* __restrict__ cnt, float* __restrict__ ts,
                        u32* __restrict__ ti, float4* __restrict__ tb, float* __restrict__ ar) {
    __shared__ u64 sk[CAND_CAP];
    u32 n = *cnt; if (n > (u32)CAND_CAP) n = (u32)CAND_CAP;
    for (u32 t = threadIdx.x; t < (u32)CAND_CAP; t += 1024u)
        sk[t] = (t < n) ? ~cand[t] : ~0ull;    /* sort ascending on ~key == descending on key */
    for (u32 size = 2; size <= (u32)CAND_CAP; size <<= 1) {
        for (u32 stride = size >> 1; stride >= 1; stride >>= 1) {
            __syncthreads();
            for (u32 t = threadIdx.x; t < (u32)(CAND_CAP / 2); t += 1024u) {
                u32 i = 2u * t - (t & (stride - 1u));
                u32 j = i + stride;
                bool asc = ((i & size) == 0u);
                u64 a = sk[i], b = sk[j];
                if ((a > b) == asc) { sk[i] = b; sk[j] = a; }
            }
        }
    }
    __syncthreads();
    for (u32 t = threadIdx.x; t < (u32)N_TOP; t += 1024u) {
        u64 kk = ~sk[t];
        float  score; u32 idx;
        if (kk == 0ull) { score = -1.0f; idx = 0u; }
        else { score = __uint_as_float((u32)(kk >> 32)); idx = 0xFFFFFFFFu - (u32)kk; }
        float4 b = boxes[idx];
        ts[t] = score; ti[t] = idx; tb[t] = b;
        ar[t] = fmaxf(b.z - b.x, 0.0f) * fmaxf(b.w - b.y, 0.0f);
    }
}

/* ---------- pass 5: pairwise IoU -> suppression bitmask ----------
 * Block = 8 waves, all sharing one 16-row tile; per-block the 16 row boxes + areas (320B)
 * are DMA'd into LDS once with gfx1250 async LDS-copy (ASYNCcnt), then each wave computes a
 * 16x32 tile. area_i + area_j comes from the matrix unit:
 *   A (16x4) = [area_row | 1 | 0 | 0], B (4x16) = [1 ; area_col ; 0 ; 0] => C = area_i + area_j
 * C lane layout (lane 0-15: M=r, N=lane; lane 16-31: M=r+8, N=lane-16) feeds the per-pair
 * intersection test; iou>T done division-free; bits packed with wave32 ballot; each mask
 * u32 written exactly once (no atomics, no mask zero-init needed). */
__global__ __launch_bounds__(256)
void iou_mask_kernel(const float4* __restrict__ tb, const float* __restrict__ ar,
                     u32* __restrict__ mask) {
    __shared__ float4 srow[16];
    __shared__ float  sarea[16];

    int rt = (int)(blockIdx.x >> 4);                 /* 0..255 row tiles (16 blocks share each) */
    int r0 = rt << 4;
    int waveId = (int)(threadIdx.x >> 5);
    int lane   = (int)(threadIdx.x & 31u);
    int ct = ((int)(blockIdx.x & 15u)) * 8 + waveId; /* 0..127 col tiles */
    int c0 = ct << 5;

    /* stage row boxes (16x16B) + row areas (64B) into LDS: 20 lanes x b128 async DMA */
    if (threadIdx.x < 20u) {
        int l = (int)threadIdx.x;
        u64 gaddr;
        u32 laddr;
        if (l < 16) {
            gaddr = (u64)(uintptr_t)(tb + r0 + l);
            laddr = (u32)(uintptr_t)(&srow[l]);
        } else {
            gaddr = (u64)(uintptr_t)(ar + r0 + (l - 16) * 4);
            laddr = (u32)(uintptr_t)(&sarea[(l - 16) * 4]);
        }
        asm volatile("global_load_async_to_lds_b128 %0, %1, off"
                     :: "v"(laddr), "v"(gaddr) : "memory");
        asm volatile("s_wait_asynccnt 0x0" ::: "memory");
    }
    __syncthreads();

    int half = lane >> 4, lm = lane & 15;

    /* A operand: lanes 0-15 carry K=0,1; lanes 16-31 carry K=2,3 (zeros) */
    v2f A;
    A.x = (half == 0) ? sarea[lm] : 0.0f;
    A.y = (half == 0) ? 1.0f      : 0.0f;

    float4 cb0 = tb[c0 + lm];
    float4 cb1 = tb[c0 + 16 + lm];
    float  ca0 = ar[c0 + lm];
    float  ca1 = ar[c0 + 16 + lm];

    /* B operand: VGPR0 lanes 0-15 = row K=0 (ones), lanes 16-31 = row K=1 (areas) */
    v2f B0; B0.x = (half == 0) ? 1.0f : ca0; B0.y = 0.0f;
    v2f B1; B1.x = (half == 0) ? 1.0f : ca1; B1.y = 0.0f;

    v8f Z = {};
    v8f S0 = __builtin_amdgcn_wmma_f32_16x16x4_f32(false, A, false, B0, (short)0, Z, false, false);
    v8f S1 = __builtin_amdgcn_wmma_f32_16x16x4_f32(false, A, false, B1, (short)0, Z, false, false);

    #pragma unroll
    for (int r = 0; r < 8; ++r) {
        float4 rb = srow[r + half * 8];              /* ds_load_b128 from staged tile */

        float ix0 = fminf(rb.z, cb0.z) - fmaxf(rb.x, cb0.x);
        float iy0 = fminf(rb.w, cb0.w) - fmaxf(rb.y, cb0.y);
        float in0 = fmaxf(ix0, 0.0f) * fmaxf(iy0, 0.0f);
        bool  p0  = in0 > IOU_THRESH * (S0[r] - in0 + 1e-9f);

        float ix1 = fminf(rb.z, cb1.z) - fmaxf(rb.x, cb1.x);
        float iy1 = fminf(rb.w, cb1.w) - fmaxf(rb.y, cb1.y);
        float in1 = fmaxf(ix1, 0.0f) * fmaxf(iy1, 0.0f);
        bool  p1  = in1 > IOU_THRESH * (S1[r] - in1 + 1e-9f);

        u32 b0m = (u32)__ballot((int)p0);   /* lo16: row r / cols c0..+15, hi16: row r+8 */
        u32 b1m = (u32)__ballot((int)p1);
        u32 wlo = (b0m & 0xFFFFu) | (b1m << 16);
        u32 whi = (b0m >> 16)     | (b1m & 0xFFFF0000u);
        if      (lane == 0) mask[(size_t)(r0 + r)     * MASK_W + ct] = wlo;
        else if (lane == 1) mask[(size_t)(r0 + r + 8) * MASK_W + ct] = whi;
    }
}

/* ---------- pass 6: greedy sequential sweep, one wave holds the 4096-bit sup mask ---------- */
__global__ void sweep_kernel(const float* __restrict__ ts, const u32* __restrict__ mask,
                             u32* __restrict__ keep) {
    int lane = (int)threadIdx.x;          /* 32 lanes; lane owns sup words lane*4 .. lane*4+3 */
    u32 s0 = 0, s1 = 0, s2 = 0, s3 = 0;   /* init sup = !valid */
    for (int k = 0; k < 4; ++k) {
        u32 w = 0;
        for (int b = 0; b < 32; ++b) {
            int j = (lane * 4 + k) * 32 + b;
            if (ts[j] < CONF_THRESH) w |= (1u << b);
        }
        if (k == 0) s0 = w; else if (k == 1) s1 = w; else if (k == 2) s2 = w; else s3 = w;
    }
    for (int i = 0; i < N_TOP; ++i) {
        /* hide latency of the serially-dependent row-OR load: prefetch next row */
        int ip = (i + 1 < N_TOP) ? (i + 1) : i;
        __builtin_prefetch(mask + (size_t)ip * MASK_W + lane * 4, 0, 0);

        int wi = i >> 5, owner = wi >> 2, k = wi & 3, bit = i & 31;
        u32 myw = (k == 0) ? s0 : (k == 1) ? s1 : (k == 2) ? s2 : s3;
        u32 w = __shfl(myw, owner, 32);
        bool is_keep = ((w >> bit) & 1u) == 0u;
        if (lane == 0) keep[i] = is_keep ? 1u : 0u;
        if (is_keep) {
            const u32* row = mask + (size_t)i * MASK_W + lane * 4;
            s0 |= row[0]; s1 |= row[1]; s2 |= row[2]; s3 |= row[3];
        }
    }
}

/* ---------- pass 7: emit [4096,5] rows, zero the suppressed ones ---------- */
__global__ void output_kernel(const float4* __restrict__ tb, const float* __restrict__ ts,
                              const u32* __restrict__ keep, float* __restrict__ out) {
    int t = blockIdx.x * blockDim.x + threadIdx.x;
    if (t >= N_TOP) return;
    float4 b = tb[t];
    float  s = ts[t];
    bool   k = keep[t] != 0u;
    float* o = out + (size_t)t * 5;
    o[0] = k ? b.x : 0.0f;
    o[1] = k ? b.y : 0.0f;
    o[2] = k ? b.z : 0.0f;
    o[3] = k ? b.w : 0.0f;
    o[4] = k ? s   : 0.0f;
}

extern "C" void kernel_launch(void* const* d_in, const int* in_sizes, int n_in,
                              void* d_out, int out_size, void* d_ws, size_t ws_size,
                              hipStream_t stream) {
    const float4* boxes = (const float4*)d_in[0];
    const float*  confs = (const float*)d_in[1];
    int n = in_sizes[1];                       /* 200000 anchors */

    uint8_t* ws   = (uint8_t*)d_ws;
    u32*    hist  = (u32*)(ws + OFF_HIST);     /* zero_kernel also clears CNT+TBIN right after */
    u32*    cnt   = (u32*)(ws + OFF_CNT);
    u32*    tbin  = (u32*)(ws + OFF_TBIN);
    u64*    cand  = (u64*)(ws + OFF_CAND);
    float*  ts    = (float*)(ws + OFF_SCORE);
    u32*    ti    = (u32*)(ws + OFF_IDX);
    float4* tb    = (float4*)(ws + OFF_BOXES);
    float*  ar    = (float*)(ws + OFF_AREA);
    u32*    keep  = (u32*)(ws + OFF_KEEP);
    u32*    mask  = (u32*)(ws + OFF_MASK);

    int blocksN = (n + 255) / 256;

    zero_kernel   <<<(4098 + 255) / 256, 256, 0, stream>>>(hist);
    hist_kernel   <<<blocksN, 256, 0, stream>>>(confs, hist, n);
    scan_kernel   <<<1, 1, 0, stream>>>(hist, tbin);
    compact_kernel<<<blocksN, 256, 0, stream>>>(confs, tbin, cnt, cand, n);
    sort_gather_kernel<<<1, 1024, 0, stream>>>(boxes, cand, cnt, ts, ti, tb, ar);
    /* 256 row-tiles x 16 blocks, each block = 8 waves = 8 col-tiles of 32 cols */
    iou_mask_kernel<<<4096, 256, 0, stream>>>(tb, ar, mask);
    sweep_kernel  <<<1, 32, 0, stream>>>(ts, mask, keep);
    output_kernel <<<(N_TOP + 255) / 256, 256, 0, stream>>>(tb, ts, keep, (float*)d_out);
}